// BiLSTMEncoder_84421877170778
// MI455X (gfx1250) — compile-verified
//
#include <hip/hip_runtime.h>
#include <math.h>

// BiLSTM: B=32, T=512, I=512, H=512, fp32 WMMA + async global->LDS staging.
#define BB    32      // batch
#define TT    512     // sequence length
#define II    512     // input size
#define HHH   512     // hidden size
#define NWG   32      // workgroups per direction
#define HPW   16      // hidden units owned per workgroup
#define KC    128     // K-chunk staged in LDS per double-buffer slot
#define NCH   8       // chunks per step: 4 x-chunks + 4 h-chunks (K = 1024)
#define APAD  132     // padded LDS row stride (floats); 132 % 64 == 4 -> conflict-free b64 reads

typedef float v2f __attribute__((ext_vector_type(2)));
typedef float v8f __attribute__((ext_vector_type(8)));

__global__ void bilstm_init_ws(unsigned* cnt) {
    cnt[threadIdx.x] = 0u;   // workspace is poisoned 0xAA by the harness
}

__device__ __forceinline__ float sigmoid_f(float v) {
    return 1.0f / (1.0f + __expf(-v));
}

// CDNA5 async copy: memory -> LDS, per-lane 16B, tracked by ASYNCcnt.
__device__ __forceinline__ void async_b128_to_lds(unsigned lds_addr, const void* gaddr) {
    asm volatile("global_load_async_to_lds_b128 %0, %1, off"
                 :: "v"(lds_addr), "v"(gaddr) : "memory");
}

__device__ __forceinline__ void wait_async0() {
#if defined(__has_builtin)
#if __has_builtin(__builtin_amdgcn_s_wait_asynccnt)
    __builtin_amdgcn_s_wait_asynccnt(0);
#else
    asm volatile("s_wait_asynccnt 0x0" ::: "memory");
#endif
#else
    asm volatile("s_wait_asynccnt 0x0" ::: "memory");
#endif
}

__global__ __launch_bounds__(256)
void bilstm_persistent(const float* __restrict__ x,      // [B,T,I]
                       const float* __restrict__ h0,     // [2,B,H]
                       const float* __restrict__ c0,     // [2,B,H]
                       const float* __restrict__ w_ih_f, const float* __restrict__ w_hh_f,
                       const float* __restrict__ b_ih_f, const float* __restrict__ b_hh_f,
                       const float* __restrict__ w_ih_r, const float* __restrict__ w_hh_r,
                       const float* __restrict__ b_ih_r, const float* __restrict__ b_hh_r,
                       float* __restrict__ out,          // [B,T,2H]
                       float* __restrict__ h_n,          // [2,B,H]
                       float* __restrict__ c_n,          // [2,B,H]
                       unsigned* __restrict__ cnt,       // [2] per-direction barrier counters
                       float* __restrict__ hbuf)         // [2 dir][2 buf][B*H]
{
    const int dir = blockIdx.x / NWG;         // 0 = forward, 1 = reverse
    const int wg  = blockIdx.x % NWG;
    const int hb  = wg * HPW;

    const float* w_ih = dir ? w_ih_r : w_ih_f;
    const float* w_hh = dir ? w_hh_r : w_hh_f;
    const float* b_ih = dir ? b_ih_r : b_ih_f;
    const float* b_hh = dir ? b_hh_r : b_hh_f;

    unsigned* bar = cnt + dir;
    float* hb0 = hbuf + (size_t)dir * 2 * BB * HHH;
    float* hb1 = hb0 + BB * HHH;

    // LDS staging (async dest needs 16B alignment). Total ~109 KB.
    __shared__ __align__(16) float alds[2][BB][APAD];   // A = [x_t ; h] chunk, 32 x 128
    __shared__ __align__(16) float blds[2][64][APAD];   // B = weight rows chunk, 64 x 128
    __shared__ float lds_g[4][BB][HPW];                 // i,f,g,o gate tiles
    __shared__ float lds_c[BB][HPW];                    // persistent cell-state slice

    const int tid  = threadIdx.x;
    const int wv   = tid >> 5;
    const int lane = tid & 31;
    const int ln   = lane & 15;           // tile col (N) / A row (M) within 16
    const int hi   = lane >> 4;           // lane-half: K pair for A/B, row+8 for C/D
    const int mt   = wv & 1;              // batch tile 0-15 / 16-31
    const int gate = wv >> 1;             // 0..3 = i,f,g,o
    const int m    = mt * 16 + ln;        // batch row for A reads
    const int gcol = gate * HHH + hb + ln;

    // --- Init h/c slice from h0/c0 ---
    for (int e = tid; e < BB * HPW; e += 256) {
        const int bbx = e >> 4, n = e & 15;
        hb0[bbx * HHH + hb + n] = h0[(size_t)dir * BB * HHH + bbx * HHH + hb + n];
        lds_c[bbx][n]           = c0[(size_t)dir * BB * HHH + bbx * HHH + hb + n];
    }

    unsigned bar_target = 0;
    auto gridbar = [&]() {
        bar_target += NWG;
        __threadfence();
        __syncthreads();
        if (tid == 0) {
            __hip_atomic_fetch_add(bar, 1u, __ATOMIC_RELEASE, __HIP_MEMORY_SCOPE_AGENT);
            while (__hip_atomic_load(bar, __ATOMIC_ACQUIRE, __HIP_MEMORY_SCOPE_AGENT) < bar_target)
                __builtin_amdgcn_s_sleep(1);
        }
        __syncthreads();
        __threadfence();
    };

    gridbar();   // h0 slices visible before t=0

    const float bias = b_ih[gcol] + b_hh[gcol];

    // Per-thread async-copy assignments (fixed for all chunks):
    const int arow = tid >> 3, aseg = tid & 7;          // A: 32 rows x 8 segs, 1/thread
    // B: 64 rows x 8 segs, 2/thread (idx = tid, tid+256)
    for (int t = 0; t < TT; ++t) {
        const int tx = dir ? (TT - 1 - t) : t;
        const float* hcur = (t & 1) ? hb1 : hb0;
        float*       hnxt = (t & 1) ? hb0 : hb1;

        // Issue one K-chunk (c in [0,8)): c<4 from x/w_ih, c>=4 from h/w_hh.
        auto issue_chunk = [&](int c) {
            const int buf = c & 1;
            {   // A chunk: rows = batch, 128 K-columns
                const float* src = (c < 4)
                    ? x    + ((size_t)arow * TT + tx) * II + c * KC + aseg * 4
                    : hcur + (size_t)arow * HHH + (c - 4) * KC + aseg * 4;
                async_b128_to_lds((unsigned)(size_t)&alds[buf][arow][aseg * 4], src);
            }
            #pragma unroll
            for (int u = 0; u < 2; ++u) {   // B chunk: 64 gate rows
                const int idx  = tid + u * 256;
                const int row  = idx >> 3, seg = idx & 7;
                const int grow = (row >> 4) * HHH + hb + (row & 15);
                const float* src = (c < 4)
                    ? w_ih + (size_t)grow * II + c * KC + seg * 4
                    : w_hh + (size_t)grow * HHH + (c - 4) * KC + seg * 4;
                async_b128_to_lds((unsigned)(size_t)&blds[buf][row][seg * 4], src);
            }
        };

        v8f acc;
        #pragma unroll
        for (int r = 0; r < 8; ++r) acc[r] = bias;   // bias is column-only -> broadcast

        issue_chunk(0);
        for (int c = 0; c < NCH; ++c) {
            wait_async0();        // my chunk-c transfers landed in LDS
            __syncthreads();      // everyone's landed; chunk c-1 compute finished
            if (c < NCH - 1)
                issue_chunk(c + 1);   // overlaps with compute below (safe: buffer free)

            const int buf = c & 1;
            #pragma unroll 4
            for (int kk = 0; kk < KC; kk += 4) {
                v2f a = *(const v2f*)&alds[buf][m][kk + 2 * hi];
                v2f b = *(const v2f*)&blds[buf][gate * 16 + ln][kk + 2 * hi];
                acc = __builtin_amdgcn_wmma_f32_16x16x4_f32(false, a, false, b,
                                                            (short)0, acc, false, false);
            }
        }

        // Scatter this wave's 16x16 gate tile to LDS. C/D row = r + 8*laneHi.
        #pragma unroll
        for (int r = 0; r < 8; ++r)
            lds_g[gate][mt * 16 + r + 8 * hi][ln] = acc[r];
        __syncthreads();

        // Elementwise LSTM cell on the owned 32x16 slice (2 elements/thread).
        for (int e = tid; e < BB * HPW; e += 256) {
            const int bbx = e >> 4, n = e & 15;
            const float ig = sigmoid_f(lds_g[0][bbx][n]);
            const float fg = sigmoid_f(lds_g[1][bbx][n]);
            const float gg = tanhf(lds_g[2][bbx][n]);
            const float og = sigmoid_f(lds_g[3][bbx][n]);
            const float cv = fg * lds_c[bbx][n] + ig * gg;
            const float hv = og * tanhf(cv);
            lds_c[bbx][n] = cv;
            hnxt[(size_t)bbx * HHH + hb + n] = hv;
            __builtin_nontemporal_store(hv,
                out + ((size_t)bbx * TT + tx) * (2 * HHH) + (size_t)dir * HHH + hb + n);
            if (t == TT - 1) {
                h_n[(size_t)dir * BB * HHH + (size_t)bbx * HHH + hb + n] = hv;
                c_n[(size_t)dir * BB * HHH + (size_t)bbx * HHH + hb + n] = cv;
            }
        }
        gridbar();   // publish h(t+1) to all 32 WGs of this direction
    }
}

extern "C" void kernel_launch(void* const* d_in, const int* in_sizes, int n_in,
                              void* d_out, int out_size, void* d_ws, size_t ws_size,
                              hipStream_t stream) {
    const float* x      = (const float*)d_in[0];
    const float* h0     = (const float*)d_in[1];
    const float* c0     = (const float*)d_in[2];
    const float* w_ih_f = (const float*)d_in[3];
    const float* w_hh_f = (const float*)d_in[4];
    const float* b_ih_f = (const float*)d_in[5];
    const float* b_hh_f = (const float*)d_in[6];
    const float* w_ih_r = (const float*)d_in[7];
    const float* w_hh_r = (const float*)d_in[8];
    const float* b_ih_r = (const float*)d_in[9];
    const float* b_hh_r = (const float*)d_in[10];

    float* out = (float*)d_out;                         // [B,T,2H]
    float* h_n = out + (size_t)BB * TT * 2 * HHH;       // [2,B,H]
    float* c_n = h_n + 2 * BB * HHH;                    // [2,B,H]

    // Workspace: [0,256) barrier counters, then 2 dir x 2 buf x B*H floats (256 KB).
    unsigned* cnt  = (unsigned*)d_ws;
    float*    hbuf = (float*)((char*)d_ws + 256);

    bilstm_init_ws<<<1, 64, 0, stream>>>(cnt);
    bilstm_persistent<<<2 * NWG, 256, 0, stream>>>(
        x, h0, c0, w_ih_f, w_hh_f, b_ih_f, b_hh_f,
        w_ih_r, w_hh_r, b_ih_r, b_hh_r,
        out, h_n, c_n, cnt, hbuf);
}